// yoloLoss_36971078484490
// MI455X (gfx1250) — compile-verified
//
#include <hip/hip_runtime.h>

// ---------------- problem constants ----------------
#define SGRID 14
#define CELLS_PER_BATCH (SGRID * SGRID)     // 196
#define CH 30                               // 10 box + 20 class
#define NBATCH 4096
#define TOTAL_CELLS (NBATCH * CELLS_PER_BATCH) // 802816
#define BLOCK 256
#define NBLOCKS (TOTAL_CELLS / BLOCK)       // 3136 (exact)
#define NV4 (BLOCK * CH / 4)                // 1920 float4 per tensor per block

__constant__ float kStep = 1.0f / 14.0f;

typedef __attribute__((ext_vector_type(2))) float v2f;
typedef __attribute__((ext_vector_type(8))) float v8f;
typedef __attribute__((ext_vector_type(4))) int v4i;

#if defined(__HIP_DEVICE_COMPILE__)
typedef __attribute__((address_space(1))) v4i* gv4i_p;  // global
typedef __attribute__((address_space(3))) v4i* lv4i_p;  // LDS
#endif

// ---- wave-level sum via V_WMMA_F32_16X16X4_F32 (exact f32, deterministic) ----
// A(16x4): lane m vgpr0 = acc (K=0 for lanes 0-15, K=2 for lanes 16-31), vgpr1 = 0.
// B(4x16) = ones  =>  D[m][n] = acc[m] + acc[m+16] for every n.
// Per-lane sum of the 8 D vgprs gives rows 0-7 (lanes 0-15) / rows 8-15 (lanes 16-31);
// one xor-16 shuffle completes the 32-lane total in every lane.
__device__ inline float wave_sum(float v) {
#if __has_builtin(__builtin_amdgcn_wmma_f32_16x16x4_f32)
  v2f a = {v, 0.0f};
  v2f b = {1.0f, 1.0f};
  v8f c = {};
  c = __builtin_amdgcn_wmma_f32_16x16x4_f32(false, a, false, b, (short)0, c,
                                            false, false);
  float s = ((c[0] + c[1]) + (c[2] + c[3])) + ((c[4] + c[5]) + (c[6] + c[7]));
  s += __shfl_xor(s, 16, 32);
  return s;
#else
  for (int off = 16; off > 0; off >>= 1) v += __shfl_xor(v, off, 32);
  return v;
#endif
}

// ---------------- kernel 1: per-block partial loss ----------------
__global__ __launch_bounds__(BLOCK) void yolo_partial(
    const float* __restrict__ pred, const float* __restrict__ targ,
    float* __restrict__ partial) {
  __shared__ float ldsP[BLOCK * CH];   // 30 KB
  __shared__ float ldsT[BLOCK * CH];   // 30 KB
  __shared__ float wsum[BLOCK / 32];

  const int tid = threadIdx.x;
  const long long base = (long long)blockIdx.x * (BLOCK * CH);
  const float4* gp = (const float4*)(pred + base);
  const float4* gt = (const float4*)(targ + base);
  float4* lp = (float4*)ldsP;
  float4* lt = (float4*)ldsT;

  // ---- coalesced global -> LDS staging (async-tensor path when available) ----
#if __has_builtin(__builtin_amdgcn_global_load_async_to_lds_b128)
  for (int i = tid; i < NV4; i += BLOCK) {
    __builtin_amdgcn_global_load_async_to_lds_b128(
        (gv4i_p)(v4i*)(gp + i), (lv4i_p)(v4i*)(lp + i), 0, 0);
    __builtin_amdgcn_global_load_async_to_lds_b128(
        (gv4i_p)(v4i*)(gt + i), (lv4i_p)(v4i*)(lt + i), 0, 0);
  }
#if __has_builtin(__builtin_amdgcn_s_wait_asynccnt)
  __builtin_amdgcn_s_wait_asynccnt(0);
#else
  asm volatile("s_wait_asynccnt 0" ::: "memory");
#endif
#else
  for (int i = tid; i < NV4; i += BLOCK) {
    lp[i] = gp[i];
    lt[i] = gt[i];
  }
#endif
  __syncthreads();

  // ---- per-cell loss (one cell per thread; LDS reads are bank-conflict-free) ----
  const float* p = ldsP + tid * CH;
  const float* t = ldsT + tid * CH;
  const int gid = blockIdx.x * BLOCK + tid;
  const int rem = gid % CELLS_PER_BATCH;
  const float gj = (float)(rem / SGRID);  // axis 1
  const float gi = (float)(rem % SGRID);  // axis 2

  float ciouv[2], confse[2], xyl[2], whl[2];
  bool o0[2];

#pragma unroll
  for (int q = 0; q < 2; ++q) {
    const float* pb = p + q * 5;
    const float* tb = t + q * 5;
    // _convert_box (clamped at 0)
    float px = fmaxf((pb[0] + gi) * kStep - pb[2] * 0.5f, 0.0f);
    float py = fmaxf((pb[1] + gj) * kStep - pb[3] * 0.5f, 0.0f);
    float pw = fmaxf(pb[2], 0.0f), ph = fmaxf(pb[3], 0.0f);
    float tx = fmaxf((tb[0] + gi) * kStep - tb[2] * 0.5f, 0.0f);
    float ty = fmaxf((tb[1] + gj) * kStep - tb[3] * 0.5f, 0.0f);
    float tw = fmaxf(tb[2], 0.0f), th = fmaxf(tb[3], 0.0f);
    // _ciou
    float x1min = px - pw * 0.5f, x1max = px + pw * 0.5f;
    float y1min = py - ph * 0.5f, y1max = py + ph * 0.5f;
    float x2min = tx - tw * 0.5f, x2max = tx + tw * 0.5f;
    float y2min = ty - th * 0.5f, y2max = ty + th * 0.5f;
    float iw = fmaxf(0.0f, fminf(x1max, x2max) - fmaxf(x1min, x2min));
    float ih = fmaxf(0.0f, fminf(y1max, y2max) - fmaxf(y1min, y2min));
    float inter = iw * ih;
    float uni = pw * ph + tw * th - inter;
    float dcx = tx - px, dcy = ty - py;
    float dc = dcx * dcx + dcy * dcy;
    float e1x = x1max - x1min, e1y = y1max - y1min;
    float e2x = x2max - x2min, e2y = y2max - y2min;
    float diag = fmaxf(0.0f, e1x * e1x + e1y * e1y) +
                 fmaxf(0.0f, e2x * e2x + e2y * e2y);
    ciouv[q] = inter / uni - dc / diag;
    // per-box loss pieces
    o0[q] = tb[4] > 0.0f;
    float d4 = pb[4] - tb[4];
    confse[q] = d4 * d4;
    float d0 = pb[0] - tb[0], d1 = pb[1] - tb[1];
    xyl[q] = d0 * d0 + d1 * d1;
    float s2 = sqrtf(tb[2]) - sqrtf(pb[2]);
    float s3 = sqrtf(tb[3]) - sqrtf(pb[3]);
    whl[q] = s2 * s2 + s3 * s3;
  }

  const bool sig = o0[1];
  const int best = (ciouv[1] > ciouv[0]) ? 1 : 0;  // argmax, first index on ties

  float clsum = 0.0f;
#pragma unroll
  for (int k = 10; k < 30; ++k) {
    float d = p[k] - t[k];
    clsum += d * d;
  }

  float loss = sig ? clsum : 0.0f;
#pragma unroll
  for (int q = 0; q < 2; ++q) {
    bool objq = sig ? (o0[q] && (q == best)) : o0[q];
    float m = objq ? 1.0f : 0.0f;
    loss += m * confse[q] + 0.5f * (1.0f - m) * confse[q] +
            5.0f * m * (xyl[q] + whl[q]);
  }

  // ---- deterministic block reduction: WMMA wave sums -> LDS -> scalar ----
  float ws = wave_sum(loss);
  if ((tid & 31) == 0) wsum[tid >> 5] = ws;
  __syncthreads();
  if (tid == 0) {
    float s = 0.0f;
#pragma unroll
    for (int i = 0; i < BLOCK / 32; ++i) s += wsum[i];
    partial[blockIdx.x] = s;
  }
}

// ---------------- kernel 2: final reduction of 3136 partials ----------------
__global__ __launch_bounds__(256) void yolo_final(
    const float* __restrict__ partial, float* __restrict__ out) {
  __shared__ float wsum[8];
  float acc = 0.0f;
  for (int i = threadIdx.x; i < NBLOCKS; i += 256) acc += partial[i];
  float ws = wave_sum(acc);
  if ((threadIdx.x & 31) == 0) wsum[threadIdx.x >> 5] = ws;
  __syncthreads();
  if (threadIdx.x == 0) {
    float s = 0.0f;
#pragma unroll
    for (int i = 0; i < 8; ++i) s += wsum[i];
    out[0] = s * (1.0f / (float)NBATCH);
  }
}

// ---------------- entry point ----------------
extern "C" void kernel_launch(void* const* d_in, const int* in_sizes, int n_in,
                              void* d_out, int out_size, void* d_ws,
                              size_t ws_size, hipStream_t stream) {
  (void)in_sizes; (void)n_in; (void)out_size; (void)ws_size;
  const float* pred = (const float*)d_in[0];
  const float* targ = (const float*)d_in[1];
  float* partial = (float*)d_ws;  // NBLOCKS floats = 12.25 KB scratch

  yolo_partial<<<NBLOCKS, BLOCK, 0, stream>>>(pred, targ, partial);
  yolo_final<<<1, 256, 0, stream>>>(partial, (float*)d_out);
}